// PureBosonicExt_35098472743596
// MI455X (gfx1250) — compile-verified
//
#include <hip/hip_runtime.h>

typedef __attribute__((ext_vector_type(2))) float v2f;
typedef __attribute__((ext_vector_type(2))) int   v2i;
typedef __attribute__((ext_vector_type(8))) float v8f;

#define SLICES 128   // K-slices per (i,j) pair -> 36*128 = 4608 waves

static __device__ __forceinline__ v8f wmma4(v2f A, v2f B, v8f C) {
  // D = A(16x4,f32) * B(4x16,f32) + C(16x16,f32), wave32
  return __builtin_amdgcn_wmma_f32_16x16x4_f32(
      /*neg_a=*/false, A, /*neg_b=*/false, B,
      /*c_mod=*/(short)0, C, /*reuse_a=*/false, /*reuse_b=*/false);
}

// ---------------------------------------------------------------------------
// ws[0] = sum(theta^2), ws[1] = raw bilinear sum
// ---------------------------------------------------------------------------
__global__ void zero_ws_kernel(float* ws) {
  ws[0] = 0.0f;
  ws[1] = 0.0f;
}

__global__ void sumsq_kernel(const float* __restrict__ theta, int n,
                             float* __restrict__ ws) {
  int tid = blockIdx.x * blockDim.x + threadIdx.x;
  int stride = gridDim.x * blockDim.x;
  float s = 0.0f;
  for (int k = tid; k < n; k += stride) {
    float v = theta[k];
    s += v * v;
  }
  #pragma unroll
  for (int off = 16; off > 0; off >>= 1) s += __shfl_xor(s, off, 32);
  if ((threadIdx.x & 31) == 0) atomicAdd(&ws[0], s);
}

// ---------------------------------------------------------------------------
// Main kernel: one wave per block; wave owns one (i,j) pair + one K-slice.
//
// 16x16 fragment row/col map: 0..5 = x, 6..11 = y, 12..15 = don't-care
// (padding & cross-block garbage is annihilated by zero weights in the
// epilogue, so no masking is needed in the hot loop).
//
// A[m,k] = theta[m*ND + idx0[r_k]],  B[k,n] = coeff[r_k]*theta[n*ND + idx1[r_k]]
// One fp32 WMMA per 4 r; all theta gathers are SGPR-base + 32-bit lane offset
// (12*ND floats < 2^31), streams walk uniform pointers with lane-const koff.
// ---------------------------------------------------------------------------
template <bool VEC>
__global__ void __launch_bounds__(32)
dicke_rho_wmma_kernel(const float* __restrict__ theta,
                      const float* __restrict__ op,
                      const int*   __restrict__ idx0,
                      const int*   __restrict__ idx1,
                      const float* __restrict__ coeff,
                      int ND, int R,
                      float* __restrict__ ws) {
  const int pair  = blockIdx.x % 36;
  const int slice = blockIdx.x / 36;
  const int i = pair / 6;
  const int j = pair % 6;

  const long long poff = (long long)pair * (long long)R;
  const int*   i0p = idx0  + poff;
  const int*   i1p = idx1  + poff;
  const float* cfp = coeff + poff;

  // K-slice bounds (multiple of 4, uniform across the wave)
  int per  = ((R + SLICES - 1) / SLICES + 3) & ~3;
  int rBeg = slice * per;
  int rEnd = min(R, rBeg + per);

  const int lane = (int)threadIdx.x;
  const int row  = lane & 15;     // M for A-frag, N for B/C-frag
  const int hi   = lane >> 4;     // 0: K=0,1 & rows 0..7 ; 1: K=2,3 & rows 8..15
  const int koff = hi << 1;       // per-lane K offset within a 4-chunk
  const unsigned toffu = (unsigned)((row < 12 ? row : 11) * ND);  // 32-bit base

  // Weights matching the C/D fragment layout: c[v] = S[M = v + 8*hi, N = row]
  float w[8];
  #pragma unroll
  for (int v = 0; v < 8; ++v) {
    int mm = v + (hi ? 8 : 0);
    int nn = row;
    float ww = 0.0f;
    if (nn < 6 && mm < 6)                                    // x-block
      ww = op[(nn * 6 + j) * 36 + (mm * 6 + i)];
    else if (nn >= 6 && nn < 12 && mm >= 6 && mm < 12)       // y-block
      ww = op[((nn - 6) * 6 + j) * 36 + ((mm - 6) * 6 + i)];
    w[v] = ww;
  }

  v8f acc0 = {};
  v8f acc1 = {};
  int base = rBeg;
  const int*   p0 = i0p + rBeg;
  const int*   p1 = i1p + rBeg;
  const float* pc = cfp + rBeg;

  // ---- steady state: 8 r per iteration, 2 independent WMMAs, no masks ----
  if (VEC) {
    for (; base + 8 <= rEnd; base += 8, p0 += 8, p1 += 8, pc += 8) {
      v2i ia0 = *(const v2i*)(p0 + koff);      // koff even -> 8B aligned (R even)
      v2i ia1 = *(const v2i*)(p0 + koff + 4);
      v2i ib0 = *(const v2i*)(p1 + koff);
      v2i ib1 = *(const v2i*)(p1 + koff + 4);
      v2f cc0 = *(const v2f*)(pc + koff);
      v2f cc1 = *(const v2f*)(pc + koff + 4);
      if (base + 2064 <= R) {    // uniform guard: stay inside the allocation
        __builtin_prefetch(p0 + 2048, 0, 1);
        __builtin_prefetch(p1 + 2048, 0, 1);
        __builtin_prefetch(pc + 2048, 0, 1);
      }
      v2f A0, B0, A1, B1;
      A0.x = theta[toffu + (unsigned)ia0.x];
      A0.y = theta[toffu + (unsigned)ia0.y];
      B0.x = cc0.x * theta[toffu + (unsigned)ib0.x];
      B0.y = cc0.y * theta[toffu + (unsigned)ib0.y];
      A1.x = theta[toffu + (unsigned)ia1.x];
      A1.y = theta[toffu + (unsigned)ia1.y];
      B1.x = cc1.x * theta[toffu + (unsigned)ib1.x];
      B1.y = cc1.y * theta[toffu + (unsigned)ib1.y];
      acc0 = wmma4(A0, B0, acc0);
      acc1 = wmma4(A1, B1, acc1);
    }
  } else {
    for (; base + 8 <= rEnd; base += 8, p0 += 8, p1 += 8, pc += 8) {
      int   ia0x = p0[koff],     ia0y = p0[koff + 1];
      int   ia1x = p0[koff + 4], ia1y = p0[koff + 5];
      int   ib0x = p1[koff],     ib0y = p1[koff + 1];
      int   ib1x = p1[koff + 4], ib1y = p1[koff + 5];
      float c0x  = pc[koff],     c0y  = pc[koff + 1];
      float c1x  = pc[koff + 4], c1y  = pc[koff + 5];
      v2f A0, B0, A1, B1;
      A0.x = theta[toffu + (unsigned)ia0x];
      A0.y = theta[toffu + (unsigned)ia0y];
      B0.x = c0x * theta[toffu + (unsigned)ib0x];
      B0.y = c0y * theta[toffu + (unsigned)ib0y];
      A1.x = theta[toffu + (unsigned)ia1x];
      A1.y = theta[toffu + (unsigned)ia1y];
      B1.x = c1x * theta[toffu + (unsigned)ib1x];
      B1.y = c1y * theta[toffu + (unsigned)ib1y];
      acc0 = wmma4(A0, B0, acc0);
      acc1 = wmma4(A1, B1, acc1);
    }
  }

  // ---- remainder: up to 2 chunks, r-bounds masked via zeroed coeff ----
  for (; base < rEnd; base += 4) {
    int r0 = base + koff;
    int r1 = r0 + 1;
    int r0c = min(r0, rEnd - 1);
    int r1c = min(r1, rEnd - 1);
    float m0 = (r0 < rEnd) ? 1.0f : 0.0f;
    float m1 = (r1 < rEnd) ? 1.0f : 0.0f;
    int   i0a = i0p[r0c], i0b = i0p[r1c];
    int   i1a = i1p[r0c], i1b = i1p[r1c];
    float c0  = cfp[r0c] * m0;
    float c1  = cfp[r1c] * m1;
    v2f A, B;
    A.x = theta[toffu + (unsigned)i0a];
    A.y = theta[toffu + (unsigned)i0b];
    B.x = c0 * theta[toffu + (unsigned)i1a];
    B.y = c1 * theta[toffu + (unsigned)i1b];
    acc0 = wmma4(A, B, acc0);
  }

  // ---- epilogue: contract with op weights, wave-reduce, accumulate ----
  v8f acc = acc0 + acc1;
  float partial = 0.0f;
  #pragma unroll
  for (int v = 0; v < 8; ++v) partial += acc[v] * w[v];
  #pragma unroll
  for (int off = 16; off > 0; off >>= 1) partial += __shfl_xor(partial, off, 32);
  if (lane == 0) atomicAdd(&ws[1], partial);
}

__global__ void finalize_kernel(const float* __restrict__ ws,
                                float* __restrict__ out) {
  out[0] = ws[1] / ws[0];
}

// ---------------------------------------------------------------------------
extern "C" void kernel_launch(void* const* d_in, const int* in_sizes, int n_in,
                              void* d_out, int out_size, void* d_ws, size_t ws_size,
                              hipStream_t stream) {
  const float* theta = (const float*)d_in[0];  // (2, 6*ND) fp32, x then y contiguous
  const float* op    = (const float*)d_in[1];  // (36,36) fp32
  const int*   idx0  = (const int*)d_in[2];    // (6,6,R) i32
  const int*   idx1  = (const int*)d_in[3];    // (6,6,R) i32
  const float* coeff = (const float*)d_in[4];  // (6,6,R) fp32
  float* ws  = (float*)d_ws;
  float* out = (float*)d_out;

  const int thetaN = in_sizes[0];      // 2*6*ND
  const int ND     = thetaN / 12;      // NUM_DICKE
  const int R      = in_sizes[2] / 36; // r-extent per (i,j)

  zero_ws_kernel<<<1, 1, 0, stream>>>(ws);
  sumsq_kernel<<<256, 256, 0, stream>>>(theta, thetaN, ws);
  if ((R & 1) == 0) {   // 8B-aligned stream loads valid
    dicke_rho_wmma_kernel<true><<<36 * SLICES, 32, 0, stream>>>(
        theta, op, idx0, idx1, coeff, ND, R, ws);
  } else {
    dicke_rho_wmma_kernel<false><<<36 * SLICES, 32, 0, stream>>>(
        theta, op, idx0, idx1, coeff, ND, R, ws);
  }
  finalize_kernel<<<1, 1, 0, stream>>>(ws, out);
}